// ConvexDecoder_2181843386874
// MI455X (gfx1250) — compile-verified
//
#include <hip/hip_runtime.h>
#include <hip/hip_bf16.h>

// ---------------------------------------------------------------------------
// ConvexDecoder forward on gfx1250 (MI455X).
//
// vertices (8,64,32,3) f32, smoothness (8,64) f32; outputs concatenated:
//   points(8,64*1626,3) | direction_h(8,64,1626,4) | overlap(512) |
//   retraction(512) | mean_out(8,64,3) | local_v(8,64,32,3)
//
// z = local_v(32x3) . dirs^T(3x1626) via V_WMMA_F32_16X16X4_F32
// (M=16 vertices, N=16 dirs, K=4 zero-padded). Each direction's 32 vertex
// projections land on a lane pair (l, l+16): reductions are one shfl_xor(16).
//
// Transcendental-bound workload -> raw v_log_f32/v_exp_f32 via amdgcn
// builtins (operands clipped to [1e-20,1e20], no denorm fixups needed) and
// log2 values cached between the sum pass and the gradient pass.
// ---------------------------------------------------------------------------

typedef __attribute__((ext_vector_type(2))) float v2f;
typedef __attribute__((ext_vector_type(8))) float v8f;

#define N_DIRS   1626
#define N_TILES  102          // ceil(1626/16) -> covers d in [0,1632)
#define PI_F     3.14159265358979323846f
#define PI2_F    1.57079632679489661923f
#define LBF      1e-20f
#define UBF      1e+20f
#define LOG10_2  0.30102999566398119521f   // log10(2)
#define LOG2_10  3.32192809488736234787f   // log2(10)

__device__ __forceinline__ float clampf(float x, float lo, float hi) {
    return fminf(fmaxf(x, lo), hi);
}
__device__ __forceinline__ float alog2(float x) {   // v_log_f32 (base 2)
    return __builtin_amdgcn_logf(x);
}
__device__ __forceinline__ float aexp2(float x) {   // v_exp_f32 (base 2)
    return __builtin_amdgcn_exp2f(x);
}

__global__ void __launch_bounds__(256)
convex_decoder_kernel(const float* __restrict__ vertices,   // (8,64,32,3)
                      const float* __restrict__ smoothness, // (8,64)
                      float* __restrict__ out_points,       // (8,64*1626,3)
                      float* __restrict__ out_dirh,         // (8,64,1626,4)
                      float* __restrict__ out_overlap,      // (8,64,1)
                      float* __restrict__ out_retract,      // (8,64,1)
                      float* __restrict__ out_mean,         // (8,64,3)
                      float* __restrict__ out_localv)       // (8,64,32,3)
{
    __shared__ float s_lv[32][3];
    __shared__ float s_mean[3];

    const int bp = blockIdx.x;                 // 0..511  (b*64 + p)
    const float* vbase = vertices + (size_t)bp * 96;
    const int t = threadIdx.x;

    // ---- mean over 32 vertices (3 components) -------------------------------
    if (t < 3) {
        float s = 0.f;
        #pragma unroll
        for (int v = 0; v < 32; ++v) s += vbase[v * 3 + t];
        s *= (1.0f / 32.0f);
        s_mean[t] = s;
        out_mean[(size_t)bp * 3 + t] = s;
    }
    __syncthreads();

    // ---- local_v to LDS + output --------------------------------------------
    if (t < 96) {
        const int c = t % 3;
        const float lv = vbase[t] - s_mean[c];
        s_lv[t / 3][c] = lv;
        out_localv[(size_t)bp * 96 + t] = lv;
    }
    if (t == 96) {
        out_overlap[bp] = 0.f;
        out_retract[bp] = 0.f;
    }
    const float pval = smoothness[bp];
    const float pinv = 1.0f / pval;
    const float pm1  = pval - 1.0f;
    __syncthreads();

    const int wave = t >> 5;
    const int lane = t & 31;
    const int hi   = lane >> 4;     // 0: lanes 0-15, 1: lanes 16-31
    const int lsub = lane & 15;

    // ---- A operands for WMMA f32 16x16x4 (ISA 7.12.2 layout) ----------------
    // lane m (<16):  V0 = A[m][0], V1 = A[m][1]   (x, y)
    // lane m+16:     V0 = A[m][2], V1 = A[m][3]   (z, 0-pad)
    v2f a0, a1;                      // vertices 0-15 / 16-31
    if (!hi) {
        a0[0] = s_lv[lsub][0];       a0[1] = s_lv[lsub][1];
        a1[0] = s_lv[16 + lsub][0];  a1[1] = s_lv[16 + lsub][1];
    } else {
        a0[0] = s_lv[lsub][2];       a0[1] = 0.f;
        a1[0] = s_lv[16 + lsub][2];  a1[1] = 0.f;
    }

    // ---- per-lane vertex registers for the dhdx back-projection -------------
    // C/D layout: lane<16 -> rows vb..vb+7 with vb=0 (acc0) / 16 (acc1);
    //             lane>=16 -> vb=8 / 24.
    float lvx0[8], lvy0[8], lvz0[8], lvx1[8], lvy1[8], lvz1[8];
    {
        const int vb = hi * 8;
        #pragma unroll
        for (int r = 0; r < 8; ++r) {
            lvx0[r] = s_lv[vb + r][0];
            lvy0[r] = s_lv[vb + r][1];
            lvz0[r] = s_lv[vb + r][2];
            lvx1[r] = s_lv[16 + vb + r][0];
            lvy1[r] = s_lv[16 + vb + r][1];
            lvz1[r] = s_lv[16 + vb + r][2];
        }
    }

    const float mx = s_mean[0], my = s_mean[1], mz = s_mean[2];

    // ---- direction tiles: 102 tiles of 16, round-robin over 8 waves ---------
    for (int tile = wave; tile < N_TILES; tile += 8) {
        const int d = tile * 16 + lsub;

        // direction d from the linspace/trig construction (all lanes compute
        // the full (dx,dy,dz) so no shuffle is needed at store time)
        float th1, th2;
        if (d < 1624) {
            const int i = d / 58;
            const int j = d - i * 58;
            th1 = -PI2_F + (float)(i + 1) * (PI_F / 29.0f);
            th2 = -PI_F  + (float)j       * (PI_F / 29.0f);
        } else {
            th1 = (d == 1624) ? -PI2_F : PI2_F;
            th2 = -PI_F;
        }
        const float c1 = __cosf(th1), s1 = __sinf(th1);
        const float c2 = __cosf(th2), s2 = __sinf(th2);
        float dx = c1 * c2, dy = c1 * s2, dz = s1;
        if (d >= N_DIRS) { dx = 0.f; dy = 0.f; dz = 0.f; }   // pad lanes

        // B operand (4x16, same lane split as A): lane n<16 -> (x,y),
        // lane n+16 -> (z, 0)
        v2f bm;
        bm[0] = hi ? dz : dx;
        bm[1] = hi ? 0.f : dy;

        // ---- z tile via WMMA: D(16v x 16d) = A(16v x 4) * B(4 x 16d) --------
        v8f czero = {};
        v8f acc0 = __builtin_amdgcn_wmma_f32_16x16x4_f32(
            false, a0, false, bm, (short)0, czero, false, false);
        v8f acc1 = __builtin_amdgcn_wmma_f32_16x16x4_f32(
            false, a1, false, bm, (short)0, czero, false, false);

        // ---- zm = max(z,0); zm_max over the 32 vertices ---------------------
        float zm[16];
        float zmax = 0.f;
        #pragma unroll
        for (int r = 0; r < 8; ++r) {
            zm[r]     = fmaxf(acc0[r], 0.f);
            zm[8 + r] = fmaxf(acc1[r], 0.f);
            zmax = fmaxf(zmax, fmaxf(zm[r], zm[8 + r]));
        }
        zmax = fmaxf(zmax, __shfl_xor(zmax, 16, 32));

        // ---- dynamic rescale k = 10^e (one v_exp via e*log2(10)) ------------
        const float expo = alog2(zmax) * (LOG10_2 * 1.0f) * pval; // log10(zmax)*p
        const float lk   = (expo < -20.0f) ? (-15.0f - expo) * pinv : 0.0f;
        const float e10  = clampf(ceilf(lk), 0.0f, 20.0f);
        const float lgkk = e10 * LOG2_10;          // log2(k)
        const float kk   = aexp2(lgkk);

        // ---- sum_v clip((zm*k)^p): cache lg[i]=log2(zm*k) for reuse ---------
        float lg[16];
        float ssum = 0.f;
        #pragma unroll
        for (int i = 0; i < 16; ++i) {
            const float l = alog2(zm[i]) + lgkk;   // -inf when zm==0
            lg[i] = l;
            float w = 0.f;
            if (zm[i] > 0.f) w = clampf(aexp2(pval * l), LBF, UBF);
            ssum += w;
        }
        ssum += __shfl_xor(ssum, 16, 32);
        const float h   = (ssum > 0.f)
                        ? clampf(aexp2(pinv * alog2(ssum)), LBF, UBF) : LBF;
        const float lgh = alog2(h);                // h in [1e-20,1e20]: safe

        // ---- dhdx = sum_v ratio^(p-1) * local_v, ratio = (zm*k)/h -----------
        // log2(ratio) = lg[i] - lgh ; ratio>0 <=> zm>0 (h>0 always)
        float ax = 0.f, ay = 0.f, az = 0.f;
        #pragma unroll
        for (int r = 0; r < 8; ++r) {
            {
                const float dh = (zm[r] > 0.f)
                    ? clampf(aexp2(pm1 * (lg[r] - lgh)), LBF, UBF) : LBF;
                ax += dh * lvx0[r]; ay += dh * lvy0[r]; az += dh * lvz0[r];
            }
            {
                const float dh = (zm[8 + r] > 0.f)
                    ? clampf(aexp2(pm1 * (lg[8 + r] - lgh)), LBF, UBF) : LBF;
                ax += dh * lvx1[r]; ay += dh * lvy1[r]; az += dh * lvz1[r];
            }
        }
        ax += __shfl_xor(ax, 16, 32);
        ay += __shfl_xor(ay, 16, 32);
        az += __shfl_xor(az, 16, 32);

        // h/k = exp2(lg(h) - lg(k)) but keep the plain divide (1 op, exact)
        const float hout = clampf(h / kk, -UBF, UBF);

        if (!hi && d < N_DIRS) {
            const size_t row = (size_t)bp * N_DIRS + d;
            out_points[row * 3 + 0] = ax + mx;
            out_points[row * 3 + 1] = ay + my;
            out_points[row * 3 + 2] = az + mz;
            out_dirh[row * 4 + 0] = dx;
            out_dirh[row * 4 + 1] = dy;
            out_dirh[row * 4 + 2] = dz;
            out_dirh[row * 4 + 3] = hout;
        }
    }
}

extern "C" void kernel_launch(void* const* d_in, const int* in_sizes, int n_in,
                              void* d_out, int out_size, void* d_ws, size_t ws_size,
                              hipStream_t stream) {
    const float* vertices   = (const float*)d_in[0];   // (8,64,32,3)
    const float* smoothness = (const float*)d_in[1];   // (8,64)
    // d_in[2] = pointcloud, unused by the reference outputs

    float* out = (float*)d_out;
    // flat offsets in return order
    float* points   = out;                              // 2,497,536
    float* dirh     = points   + (size_t)8 * 64 * 1626 * 3;
    float* overlap  = dirh     + (size_t)8 * 64 * 1626 * 4;
    float* retract  = overlap  + 512;
    float* meanout  = retract  + 512;
    float* localv   = meanout  + 1536;

    convex_decoder_kernel<<<dim3(512), dim3(256), 0, stream>>>(
        vertices, smoothness, points, dirh, overlap, retract, meanout, localv);
}